// SuperLoss_21921513079278
// MI455X (gfx1250) — compile-verified
//
#include <hip/hip_runtime.h>
#include <stdint.h>

typedef float v4f __attribute__((ext_vector_type(4)));

#define RED_BLOCKS   2048
#define RED_THREADS  256
#define MAP_BLOCKS   2048
#define MAP_THREADS  256
#define HALLEY_ITERS 4   // cubic convergence: 4 iters

// ---------------------------------------------------------------------------
// SuperLoss element: y = 0.5*max(-2/e, (l-tau)/0.1); w = W0(y);
// out = (l-tau)*exp(-w) + 0.1*w^2     (log(sigma) == -w exactly)
// ---------------------------------------------------------------------------
__device__ __forceinline__ float superloss_elem(float l, float tau) {
  const float NEG_TWO_OVER_E = -0.7357588823428847f;
  const float E_F            =  2.718281828459045f;
  float d = l - tau;
  float y = 0.5f * fmaxf(NEG_TWO_OVER_E, d * 10.0f);

  // Branch-point series init (y<0) and log1p init (y>=0), selected branchlessly.
  float p  = __builtin_sqrtf(fmaxf(2.0f * fmaf(E_F, y, 1.0f), 0.0f));
  float p2 = p * p;
  float wb = fmaf(11.0f / 72.0f, p2 * p, fmaf(-(1.0f / 3.0f), p2, p - 1.0f));
  float wl = __logf(1.0f + y);
  float w  = (y < 0.0f) ? wb : wl;

#pragma unroll
  for (int i = 0; i < HALLEY_ITERS; ++i) {
    float ew   = __expf(w);                       // v_exp_f32 (TRANS)
    float f    = fmaf(w, ew, -y);
    float wp1  = w + 1.0f;
    float safe = (fabsf(wp1) < 1e-6f) ? 1e-6f : wp1;
    float t    = __fdividef(f, 2.0f * safe);      // v_rcp_f32 (TRANS)
    float den  = fmaf(-(w + 2.0f), t, ew * wp1);
    den        = (fabsf(den) < 1e-12f) ? 1e-12f : den;
    w          = w - __fdividef(f, den);          // v_rcp_f32 (TRANS)
  }
  float sigma = __expf(-w);
  return fmaf(d, sigma, 0.1f * (w * w));
}

__device__ __forceinline__ v4f superloss_v4(v4f v, float tau) {
  v4f r;
  r.x = superloss_elem(v.x, tau);
  r.y = superloss_elem(v.y, tau);
  r.z = superloss_elem(v.z, tau);
  r.w = superloss_elem(v.w, tau);
  return r;
}

// ---------------------------------------------------------------------------
// Pass 3 (emitted first): elementwise map with double-buffered
// GLOBAL_LOAD_ASYNC_TO_LDS_B128 (ASYNCcnt-tracked gfx1250 async copy),
// ds_load_b128 readback (each lane reads only its own 16 bytes -> no barrier;
// s_wait_dscnt 0 guards the WAR on stage reuse), compute, non-temporal b128
// stores so the 128MB of output doesn't evict the L2-resident input.
// Grid geometry is compile-time constant so `full` is a shift, not a division.
// ---------------------------------------------------------------------------
__global__ __launch_bounds__(MAP_THREADS) void superloss_map(
    const float* __restrict__ in, float* __restrict__ out,
    const float* __restrict__ tau_p, int n) {
  __shared__ v4f stage[2][MAP_THREADS];

  const float tau = tau_p[0];
  const int tid  = threadIdx.x;
  const int gtid = (int)blockIdx.x * MAP_THREADS + tid;
  const int tot  = MAP_BLOCKS * MAP_THREADS;   // constant: 2^19 threads
  const int n4   = n >> 2;
  const int full = n4 / tot;                   // constant divisor -> shift

  // LDS byte offsets: low 32 bits of the flat shared address == LDS offset
  const unsigned lds0 = (unsigned)(uintptr_t)&stage[0][tid];
  const unsigned lds1 = (unsigned)(uintptr_t)&stage[1][tid];

  const v4f* in4  = (const v4f*)in;
  v4f*       out4 = (v4f*)out;

  if (full > 0) {
    // prologue: tile 0 -> stage 0
    asm volatile("global_load_async_to_lds_b128 %0, %1, %2"
                 :: "v"(lds0), "v"(gtid * 16), "s"(in) : "memory");
    for (int it = 0; it + 1 < full; ++it) {
      const unsigned ldsn = ((it + 1) & 1) ? lds1 : lds0;
      // WAR guard: prior ds_load of the stage we are about to overwrite is done
      asm volatile("s_wait_dscnt 0x0" ::: "memory");
      asm volatile("global_load_async_to_lds_b128 %0, %1, %2"
                   :: "v"(ldsn), "v"((gtid + (it + 1) * tot) * 16), "s"(in)
                   : "memory");
      // async loads complete in order: <=1 outstanding => tile `it` landed
      asm volatile("s_wait_asynccnt 0x1" ::: "memory");
      v4f v = stage[it & 1][tid];                     // ds_load_b128
      v4f r = superloss_v4(v, tau);
      __builtin_nontemporal_store(r, &out4[gtid + it * tot]);
    }
    asm volatile("s_wait_asynccnt 0x0" ::: "memory");
    {
      const int it = full - 1;
      v4f v = stage[it & 1][tid];
      v4f r = superloss_v4(v, tau);
      __builtin_nontemporal_store(r, &out4[gtid + it * tot]);
    }
  }
  // vector tail (non-uniform remainder of full float4 tiles)
  for (int i = full * tot + gtid; i < n4; i += tot) {
    v4f r = superloss_v4(in4[i], tau);
    __builtin_nontemporal_store(r, &out4[i]);
  }
  // scalar tail (n % 4)
  for (int i = (n4 << 2) + gtid; i < n; i += tot)
    out[i] = superloss_elem(in[i], tau);
}

// ---------------------------------------------------------------------------
// Pass 1: per-block partial sums (b128 loads populate the 192MB L2 for pass 3)
// ---------------------------------------------------------------------------
__global__ __launch_bounds__(RED_THREADS) void superloss_reduce1(
    const float* __restrict__ in, float* __restrict__ partial, int n) {
  __shared__ float sm[RED_THREADS];
  const int tid = threadIdx.x;
  const int tot = RED_BLOCKS * RED_THREADS;    // constant stride
  const int n4  = n >> 2;
  const v4f* in4 = (const v4f*)in;

  v4f acc4 = (v4f)0.0f;
  for (int i = (int)blockIdx.x * RED_THREADS + tid; i < n4; i += tot)
    acc4 += in4[i];
  float acc = (acc4.x + acc4.y) + (acc4.z + acc4.w);
  for (int i = (n4 << 2) + (int)blockIdx.x * RED_THREADS + tid; i < n; i += tot)
    acc += in[i];

  sm[tid] = acc;
  __syncthreads();
  for (int off = RED_THREADS >> 1; off > 0; off >>= 1) {
    if (tid < off) sm[tid] += sm[tid + off];
    __syncthreads();
  }
  if (tid == 0) partial[blockIdx.x] = sm[0];
}

// ---------------------------------------------------------------------------
// Pass 2: final reduce of 2048 partials -> tau = sum/N (deterministic tree)
// ---------------------------------------------------------------------------
__global__ __launch_bounds__(RED_THREADS) void superloss_reduce2(
    const float* __restrict__ partial, float* __restrict__ tau_out,
    int nparts, float inv_n) {
  __shared__ float sm[RED_THREADS];
  const int tid = threadIdx.x;
  float acc = 0.0f;
  for (int i = tid; i < nparts; i += RED_THREADS) acc += partial[i];
  sm[tid] = acc;
  __syncthreads();
  for (int off = RED_THREADS >> 1; off > 0; off >>= 1) {
    if (tid < off) sm[tid] += sm[tid + off];
    __syncthreads();
  }
  if (tid == 0) tau_out[0] = sm[0] * inv_n;
}

// ---------------------------------------------------------------------------
extern "C" void kernel_launch(void* const* d_in, const int* in_sizes, int n_in,
                              void* d_out, int out_size, void* d_ws, size_t ws_size,
                              hipStream_t stream) {
  const float* in  = (const float*)d_in[0];
  float*       out = (float*)d_out;
  const int    n   = in_sizes[0];

  float* ws       = (float*)d_ws;
  float* tau      = ws;        // ws[0]
  float* partials = ws + 64;   // 256B-offset scratch, RED_BLOCKS floats

  superloss_reduce1<<<RED_BLOCKS, RED_THREADS, 0, stream>>>(in, partials, n);
  superloss_reduce2<<<1, RED_THREADS, 0, stream>>>(partials, tau, RED_BLOCKS,
                                                   1.0f / (float)n);
  superloss_map<<<MAP_BLOCKS, MAP_THREADS, 0, stream>>>(in, out, tau, n);
}